// PtModule_76166950028102
// MI455X (gfx1250) — compile-verified
//
#include <hip/hip_runtime.h>

// out[i] = sum(x[i,:]) * i, N=65536, D=1024, fp32.
// Memory-bound (256 MB @ 23.3 TB/s ~= 11.5 us). Row-sum expressed as WMMA GEMV
// against a ones B-matrix: V_WMMA_F32_16X16X4_F32, D[m,n] = sum_k A[m,k].
// Each wave owns 16 rows; lane l feeds A with a b64 load of
// x[row0+(l&15), k + 2*(l>>4)] per K-chunk of 4 (ISA 16x4 f32 A layout).

typedef __attribute__((ext_vector_type(2))) float v2f;
typedef __attribute__((ext_vector_type(8))) float v8f;

#define D_DIM 1024
#define WAVES_PER_BLOCK 8
#define ROWS_PER_WAVE 16
#define ROWS_PER_BLOCK (WAVES_PER_BLOCK * ROWS_PER_WAVE)

static __device__ __forceinline__ v8f wmma_rowsum_acc(v2f a, v8f c) {
  const v2f ones = {1.0f, 1.0f};  // all-ones B is invariant to the B lane layout
  // 8 args: (neg_a, A, neg_b, B, c_mod, C, reuse_a, reuse_b)
  return __builtin_amdgcn_wmma_f32_16x16x4_f32(
      /*neg_a=*/false, a, /*neg_b=*/false, ones,
      /*c_mod=*/(short)0, c, /*reuse_a=*/false, /*reuse_b=*/false);
}

__global__ __launch_bounds__(256, 1) void rowsum_scale_wmma(
    const float* __restrict__ x, float* __restrict__ out, int n_rows) {
  const int lane = threadIdx.x & 31;
  const int wave = threadIdx.x >> 5;
  const long tile = (long)blockIdx.x * WAVES_PER_BLOCK + wave;
  const long row0 = tile * ROWS_PER_WAVE;
  if (row0 >= n_rows) return;  // uniform per wave

  const int m  = lane & 15;        // row within the 16-row tile
  const int kh = (lane >> 4) * 2;  // lanes 16-31 supply K=2,3 of each chunk

  if (row0 + ROWS_PER_WAVE <= (long)n_rows) {
    // ---- WMMA path: EXEC is all-ones here (no divergence before the WMMAs) ----
    const float* rowp = x + (row0 + m) * (long)D_DIM + kh;

    v8f acc0 = {0.f, 0.f, 0.f, 0.f, 0.f, 0.f, 0.f, 0.f};
    v8f acc1 = {0.f, 0.f, 0.f, 0.f, 0.f, 0.f, 0.f, 0.f};

    // 128 iterations of (2 x b64 load + 2 x wmma); unroll 4 -> 8 loads + 8 wmma
    // per body so the loads batch ahead of the dependent WMMA chain.
#pragma unroll 4
    for (int k = 0; k < D_DIM; k += 8) {
      v2f a0 = *(const v2f*)(rowp + k);      // covers columns [k,   k+3] w/ partner lane
      v2f a1 = *(const v2f*)(rowp + k + 4);  // covers columns [k+4, k+7]
      acc0 = wmma_rowsum_acc(a0, acc0);
      acc1 = wmma_rowsum_acc(a1, acc1);
    }
    v8f acc = acc0 + acc1;

    // D layout: VGPR j, lanes 0-15 -> M=j, lanes 16-31 -> M=j+8 (constant over N).
    // Lane 0 writes rows row0..row0+7, lane 16 writes rows row0+8..row0+15.
    if (m == 0) {
      long r = row0 + (long)(lane >> 4) * 8;
#pragma unroll
      for (int j = 0; j < 8; ++j) {
        long row = r + j;
        out[row] = acc[j] * (float)row;
      }
    }
  } else {
    // ---- tail tile (never taken for N=65536, kept for generality) ----
    if (lane < 16 && row0 + m < (long)n_rows) {
      const float* rp = x + (row0 + m) * (long)D_DIM;
      float s = 0.f;
      for (int k = 0; k < D_DIM; ++k) s += rp[k];
      long row = row0 + m;
      out[row] = s * (float)row;
    }
  }
}

extern "C" void kernel_launch(void* const* d_in, const int* in_sizes, int n_in,
                              void* d_out, int out_size, void* d_ws, size_t ws_size,
                              hipStream_t stream) {
  (void)in_sizes; (void)n_in; (void)d_ws; (void)ws_size;
  const float* x = (const float*)d_in[0];
  float* out = (float*)d_out;
  const int n_rows = out_size;  // N (=65536); D fixed at 1024 per reference

  const int blocks = (n_rows + ROWS_PER_BLOCK - 1) / ROWS_PER_BLOCK;  // 512
  rowsum_scale_wmma<<<blocks, 256, 0, stream>>>(x, out, n_rows);
}